// CannyEdgeLayer_23545010716914
// MI455X (gfx1250) — compile-verified
//
#include <hip/hip_runtime.h>
#include <stdint.h>

// ---------------------------------------------------------------------------
// Canny edge detection, MI455X (gfx1250).
// Memory-bound: 50.3 MB in / 50.3 MB out -> ~4.3 us HBM floor at 23.3 TB/s.
// All intermediates live in d_ws and are L2-resident (192 MB L2 >> 126 MB
// working set). Compute sits near the VALU/memory balance point (~50 ops/px),
// so the Sobel vertical passes are offloaded to the matrix pipe as banded
// 16x16x4 f32 WMMAs, and tiles (incl. wrap-around halos) are staged with the
// gfx1250 async global->LDS path (ASYNCcnt + s_wait_asynccnt).
// ---------------------------------------------------------------------------

#define BATCH 16
#define IMH   512
#define IMW   512
#define CHN   3
#define HW    (IMH * IMW)
#define PLANES (BATCH * CHN)          // 48 independent (batch, channel) planes
#define NTOT  (BATCH * IMH * IMW * CHN)

typedef __attribute__((ext_vector_type(2))) float v2f;
typedef __attribute__((ext_vector_type(8))) float v8f;

// low 32 bits of a generic pointer into __shared__ = LDS byte address
__device__ __forceinline__ uint32_t lds_addr_of(const void* p) {
  return (uint32_t)(uintptr_t)p;
}

// gfx1250 async global->LDS copy (per-lane gather), tracked by ASYNCcnt.
__device__ __forceinline__ void async_load_b32(uint32_t lds, const void* gaddr) {
  asm volatile("global_load_async_to_lds_b32 %0, %1, off"
               :: "v"(lds), "v"(gaddr) : "memory");
}
__device__ __forceinline__ void wait_async0() {
  asm volatile("s_wait_asynccnt 0x0" ::: "memory");
}

// ---------------------------------------------------------------------------
// K0: NHWC -> planar [plane][h][w]. 2D grid: no per-thread div/rem, planar
// write fully coalesced; the stride-3 NHWC gather is HBM-read-once (3 planes
// share each cacheline via L2).
// ---------------------------------------------------------------------------
__global__ void k_transpose(const float* __restrict__ x, float* __restrict__ xp) {
  int hw = blockIdx.x * 256 + threadIdx.x;        // 0 .. HW-1
  int p  = blockIdx.y;                            // plane = b*CHN + c
  int b  = p / CHN, c = p % CHN;                  // uniform -> SALU
  xp[(size_t)p * HW + hw] = x[((size_t)b * HW + hw) * CHN + c];
}

// banded vertical filters as A-matrix entries, generated from lane indices
__device__ __forceinline__ float bandS(int m, int k) {   // [1,2,1] smooth
  int d = k - m;
  return (d == 1) ? 2.0f : ((d == 0 || d == 2) ? 1.0f : 0.0f);
}
__device__ __forceinline__ float bandD(int m, int k) {   // [-1,0,+1] diff
  int d = k - m;
  return (d == 0) ? -1.0f : ((d == 2) ? 1.0f : 0.0f);
}

// ---------------------------------------------------------------------------
// K1: fused Gaussian blur + Sobel + magnitude + direction sector.
// ONE WAVE (32 threads) per 16x16 tile: WMMA needs EXEC all-ones, and
// single-wave workgroups make barriers free (-> S_NOP per ISA).
//   xs(20x20, async-staged, zero-pad) --[1 2 1]h--> hs(20x18)
//   --[1 2 1]v/16--> bl(18x18) --[-1 0 1]h / [1 2 1]h--> hx,hy(18x16)
//   gx = Sv(16x18)*hx, gy = Dv(16x18)*hy : 5+5 v_wmma_f32_16x16x4_f32,
//   A banded matrices built in registers (A layout: lane->M, vgpr+half->K),
//   accumulators in the documented f32 C/D layout (M = j + 8*(lane>=16)).
// ---------------------------------------------------------------------------
__global__ void __launch_bounds__(32)
k_front(const float* __restrict__ xp, float* __restrict__ magP,
        uint8_t* __restrict__ secP) {
  __shared__ float xs[20][20];
  __shared__ float hs[20][18];
  __shared__ float bl[18][18];
  __shared__ float hx[18][16];
  __shared__ float hy[18][16];

  const int lane = threadIdx.x;
  const int gid  = blockIdx.x;
  const int txt  = gid & 31;            // W/16 = 32 tiles
  const int tyt  = (gid >> 5) & 31;     // H/16 = 32 tiles
  const int p    = gid >> 10;
  const int h0   = tyt * 16, w0 = txt * 16;
  const float* plane = xp + (size_t)p * HW;

  for (int idx = lane; idx < 400; idx += 32) {
    int ty = idx / 20, tx = idx % 20;
    int gh = h0 - 2 + ty, gw = w0 - 2 + tx;
    if (gh >= 0 && gh < IMH && gw >= 0 && gw < IMW)
      async_load_b32(lds_addr_of(&xs[ty][tx]), plane + (size_t)gh * IMW + gw);
    else
      xs[ty][tx] = 0.0f;                             // conv 'SAME' zero pad
  }
  wait_async0();
  __syncthreads();

  for (int idx = lane; idx < 360; idx += 32) {       // 20 x 18
    int ty = idx / 18, tx = idx % 18;
    hs[ty][tx] = xs[ty][tx] + 2.0f * xs[ty][tx + 1] + xs[ty][tx + 2];
  }
  __syncthreads();
  for (int idx = lane; idx < 324; idx += 32) {       // 18 x 18
    int by = idx / 18, bx = idx % 18;
    bl[by][bx] = (hs[by][bx] + 2.0f * hs[by + 1][bx] + hs[by + 2][bx]) * (1.0f / 16.0f);
  }
  __syncthreads();
  for (int idx = lane; idx < 288; idx += 32) {       // 18 x 16
    int r = idx / 16, cidx = idx % 16;
    float b0 = bl[r][cidx], b1 = bl[r][cidx + 1], b2 = bl[r][cidx + 2];
    hx[r][cidx] = b2 - b0;                           // horizontal [-1 0 1]
    hy[r][cidx] = b0 + 2.0f * b1 + b2;               // horizontal [ 1 2 1]
  }
  __syncthreads();

  // --- matrix-pipe vertical passes: gx = Sv*hx, gy = Dv*hy (K=18 pad 20) ---
  const int hi = lane >> 4;             // lane half selects K pair / M group
  const int m  = lane & 15;             // A-matrix row
  const int n  = lane & 15;             // B/D column
  v8f accX = {}, accY = {};
#pragma unroll
  for (int c = 0; c < 5; ++c) {
    int k0 = 4 * c + 2 * hi;
    v2f aS, aD, bX, bY;
    aS.x = bandS(m, k0); aS.y = bandS(m, k0 + 1);
    aD.x = bandD(m, k0); aD.y = bandD(m, k0 + 1);
    bX.x = (k0     < 18) ? hx[k0][n]     : 0.0f;
    bX.y = (k0 + 1 < 18) ? hx[k0 + 1][n] : 0.0f;
    bY.x = (k0     < 18) ? hy[k0][n]     : 0.0f;
    bY.y = (k0 + 1 < 18) ? hy[k0 + 1][n] : 0.0f;
    accX = __builtin_amdgcn_wmma_f32_16x16x4_f32(false, aS, false, bX,
                                                 (short)0, accX, false, false);
    accY = __builtin_amdgcn_wmma_f32_16x16x4_f32(false, aD, false, bY,
                                                 (short)0, accY, false, false);
  }

  // epilogue straight out of the C/D layout: VGPR j -> (M = j+8*hi, N = n)
#pragma unroll
  for (int j = 0; j < 8; ++j) {
    float gx = accX[j], gy = accY[j];
    float mag = sqrtf(gx * gx + gy * gy);
    float ax = fabsf(gx), ay = fabsf(gy);
    uint8_t sec;
    if (ay <= 0.4142135624f * ax)      sec = 0;      // incl. gx=gy=0 -> angle 0
    else if (ay > 2.4142135624f * ax)  sec = 2;
    else                               sec = (gx * gy >= 0.0f) ? 1 : 3;
    size_t o = (size_t)p * HW + (size_t)(h0 + j + 8 * hi) * IMW + (w0 + n);
    magP[o] = mag;
    secP[o] = sec;
  }
}

// ---------------------------------------------------------------------------
// K2: NMS (wrap-around neighbors, matching jnp.roll) + double threshold.
// ---------------------------------------------------------------------------
__global__ void k_nms(const float* __restrict__ magP, const uint8_t* __restrict__ secP,
                      float* __restrict__ edges, uint8_t* __restrict__ weakP) {
  int i = blockIdx.x * 256 + threadIdx.x;
  if (i >= NTOT) return;
  int hw = i % HW;
  int po = i - hw;
  int w = hw % IMW, h = hw / IMW;
  float m = magP[i];
  int s = secP[i];
  int wm = (w == 0) ? IMW - 1 : w - 1, wp = (w == IMW - 1) ? 0 : w + 1;
  int hm = (h == 0) ? IMH - 1 : h - 1, hp = (h == IMH - 1) ? 0 : h + 1;
  int ia, ib;
  if (s == 0)      { ia = po + h * IMW + wp;  ib = po + h * IMW + wm;  }  // E/W
  else if (s == 1) { ia = po + hp * IMW + wm; ib = po + hm * IMW + wp; }  // 45
  else if (s == 2) { ia = po + hp * IMW + w;  ib = po + hm * IMW + w;  }  // N/S
  else             { ia = po + hp * IMW + wp; ib = po + hm * IMW + wm; }  // 135
  float na = magP[ia], nb = magP[ib];
  float sup = (m >= na && m >= nb) ? m : 0.0f;
  edges[i] = (sup >= 0.3f) ? 1.0f : 0.0f;
  weakP[i] = (sup >= 0.1f && sup < 0.3f) ? (uint8_t)1 : (uint8_t)0;
}

// ---------------------------------------------------------------------------
// K3 (x24): hysteresis relaxation, one 32x32 tile/block, async-staged edge
// tile + wrap halo in LDS. Monotone 0->1 updates => in-place relaxation
// converges to the unique fixed point of the reference while_loop. No-weak
// tiles exit before touching the edge map; unchanged tiles skip write-back.
// ---------------------------------------------------------------------------
__global__ void k_hyst(float* __restrict__ edges, const uint8_t* __restrict__ weakP) {
  __shared__ float ef[34][34];
  __shared__ int sAny, sChanged, sWrote;
  const int tid = threadIdx.x;
  const int gid = blockIdx.x;
  const int txt = gid & 15, tyt = (gid >> 4) & 15, p = gid >> 8;
  const int h0 = tyt * 32, w0 = txt * 32;
  const size_t po = (size_t)p * HW;
  const int tx = tid & 31;
  const int tyBase = (tid >> 5) * 4;          // 4 rows per thread

  if (tid == 0) { sAny = 0; sWrote = 0; }
  uint8_t wk[4];
#pragma unroll
  for (int k = 0; k < 4; ++k)
    wk[k] = weakP[po + (size_t)(h0 + tyBase + k) * IMW + (w0 + tx)];
  int anyw = wk[0] | wk[1] | wk[2] | wk[3];
  __syncthreads();
  if (anyw) sAny = 1;
  __syncthreads();
  if (!sAny) return;                          // nothing can change in this tile

#pragma unroll
  for (int k = 0; k < 4; ++k) {
    int ty = tyBase + k;
    async_load_b32(lds_addr_of(&ef[ty + 1][tx + 1]),
                   edges + po + (size_t)(h0 + ty) * IMW + (w0 + tx));
  }
  if (tid < 132) {                            // wrap-around 1-px halo ring
    int hy, hx2;
    if (tid < 34)       { hy = -1;        hx2 = tid - 1;   }
    else if (tid < 68)  { hy = 32;        hx2 = tid - 35;  }
    else if (tid < 100) { hy = tid - 68;  hx2 = -1;        }
    else                { hy = tid - 100; hx2 = 32;        }
    int gh = h0 + hy;  gh = (gh < 0) ? gh + IMH : ((gh >= IMH) ? gh - IMH : gh);
    int gw = w0 + hx2; gw = (gw < 0) ? gw + IMW : ((gw >= IMW) ? gw - IMW : gw);
    async_load_b32(lds_addr_of(&ef[hy + 1][hx2 + 1]),
                   edges + po + (size_t)gh * IMW + gw);
  }
  wait_async0();
  __syncthreads();

  bool wrote = false;
  for (int it = 0; it < 64; ++it) {
    if (tid == 0) sChanged = 0;
    __syncthreads();
    bool ch = false;
#pragma unroll
    for (int k = 0; k < 4; ++k) {
      int ly = tyBase + k + 1, lx = tx + 1;
      if (wk[k] && ef[ly][lx] == 0.0f) {
        float nb = ef[ly - 1][lx - 1] + ef[ly - 1][lx] + ef[ly - 1][lx + 1]
                 + ef[ly][lx - 1]                      + ef[ly][lx + 1]
                 + ef[ly + 1][lx - 1] + ef[ly + 1][lx] + ef[ly + 1][lx + 1];
        if (nb > 0.0f) { ef[ly][lx] = 1.0f; ch = true; }
      }
    }
    if (ch) { sChanged = 1; wrote = true; }
    __syncthreads();
    if (!sChanged) break;                     // uniform: read after barrier
  }
  if (wrote) sWrote = 1;
  __syncthreads();
  if (!sWrote) return;                        // tile already at fixed point

#pragma unroll
  for (int k = 0; k < 4; ++k) {
    int ty = tyBase + k;
    edges[po + (size_t)(h0 + ty) * IMW + (w0 + tx)] = ef[ty + 1][tx + 1];
  }
}

// ---------------------------------------------------------------------------
// K4: planar edge map -> NHWC output (coalesced NHWC write).
// ---------------------------------------------------------------------------
__global__ void k_pack(const float* __restrict__ edges, float* __restrict__ out) {
  int i = blockIdx.x * 256 + threadIdx.x;
  if (i >= NTOT) return;
  int c  = i % CHN;
  int t  = i / CHN;
  int w  = t % IMW;
  int t2 = t / IMW;
  int h  = t2 % IMH;
  int b  = t2 / IMH;
  out[i] = edges[(size_t)(b * CHN + c) * HW + h * IMW + w];
}

extern "C" void kernel_launch(void* const* d_in, const int* in_sizes, int n_in,
                              void* d_out, int out_size, void* d_ws, size_t ws_size,
                              hipStream_t stream) {
  (void)in_sizes; (void)n_in; (void)out_size; (void)ws_size;
  const float* x = (const float*)d_in[0];
  float* out = (float*)d_out;

  // Workspace layout (L2-resident; 10*NTOT bytes ~= 126 MB):
  //   xp   : NTOT f32  planar input, reused as the planar edge map after K1
  //   magP : NTOT f32  gradient magnitude
  //   secP : NTOT u8   direction sector (0..3)
  //   weakP: NTOT u8   weak-pixel mask
  float*   xp    = (float*)d_ws;
  float*   magP  = xp + (size_t)NTOT;
  uint8_t* secP  = (uint8_t*)(magP + (size_t)NTOT);
  uint8_t* weakP = secP + (size_t)NTOT;
  float*   edges = xp;                        // xp dead after k_front

  const int elemBlocks = NTOT / 256;          // 49152
  const int tileBlocks = PLANES * 32 * 32;    // 49152 (16x16 tiles, 1 wave each)
  const int hystBlocks = PLANES * 16 * 16;    // 12288 (32x32 tiles)

  k_transpose<<<dim3(HW / 256, PLANES), 256, 0, stream>>>(x, xp);
  k_front   <<<tileBlocks, 32, 0, stream>>>(xp, magP, secP);
  k_nms     <<<elemBlocks, 256, 0, stream>>>(magP, secP, edges, weakP);
  for (int r = 0; r < 24; ++r)
    k_hyst  <<<hystBlocks, 256, 0, stream>>>(edges, weakP);
  k_pack    <<<elemBlocks, 256, 0, stream>>>(edges, out);
}